// Diag_5299989644060
// MI455X (gfx1250) — compile-verified
//
#include <hip/hip_runtime.h>
#include <cstdint>

// Problem geometry (fixed by the reference): B=64, C=16, H=128, W=128.
// N = B*C*H*W = 16,777,216 elements per tensor; plane = H*W = 16384.
#define HW_PLANE   16384
#define BLOCK      256
#define EPT        4              // elements per thread
#define EPB        (BLOCK * EPT)  // 1024 elements per block; 1024 | 16384

// Clang native vector (required by __builtin_nontemporal_*; HIP float4 is a struct).
typedef __attribute__((ext_vector_type(4))) float v4f;

// out[complex e] = (xr[e]*s, xi[e]*s), s = exp(betas[e % 16384]).
// Each block covers 1024 consecutive elements -> one contiguous 4KB betas
// slice, staged into LDS via the CDNA5 async Global->LDS path.
__global__ __launch_bounds__(BLOCK) void diag_exp_scale_kernel(
    const float* __restrict__ xr,
    const float* __restrict__ xi,
    const float* __restrict__ betas,
    float* __restrict__ out)
{
    __shared__ float s_beta[EPB];  // 4 KB

    const int        tid   = threadIdx.x;
    const long long  base  = (long long)blockIdx.x * EPB;
    const int        slice = (int)(base & (HW_PLANE - 1));  // uniform per block

    // ---- CDNA5 async path: stage this block's 4KB betas slice into LDS ----
    // Each lane copies 16B from global address gsrc + goff to the LDS address in lds_addr; ASYNCcnt-tracked.
    {
        unsigned lds_addr = (unsigned)(uintptr_t)&s_beta[0] + (unsigned)(tid * 16);
        unsigned goff     = (unsigned)(tid * 16);
        const float* gsrc = betas + slice;  // uniform -> SGPR pair
        asm volatile("global_load_async_to_lds_b128 %0, %1, %2"
                     :
                     : "v"(lds_addr), "v"(goff), "s"(gsrc)
                     : "memory");
    }

    // ---- Streaming bulk loads (one-touch data: non-temporal b128) ----
    const v4f r  = __builtin_nontemporal_load((const v4f*)(xr + base) + tid);
    const v4f im = __builtin_nontemporal_load((const v4f*)(xi + base) + tid);

    // Wait for the async LDS fill, then make it visible.
    asm volatile("s_wait_asynccnt 0" ::: "memory");
    __syncthreads();

    const v4f b = *(const v4f*)&s_beta[tid * EPT];  // ds_load_b128
    const float s0 = __expf(b.x);
    const float s1 = __expf(b.y);
    const float s2 = __expf(b.z);
    const float s3 = __expf(b.w);

    // complex64 layout: out[2e] = re, out[2e+1] = im (interleaved).
    v4f o0, o1;
    o0.x = r.x * s0;  o0.y = im.x * s0;
    o0.z = r.y * s1;  o0.w = im.y * s1;
    o1.x = r.z * s2;  o1.y = im.z * s2;
    o1.z = r.w * s3;  o1.w = im.w * s3;

    v4f* op = (v4f*)(out + 2 * base) + 2 * tid;
    __builtin_nontemporal_store(o0, op);
    __builtin_nontemporal_store(o1, op + 1);
}

extern "C" void kernel_launch(void* const* d_in, const int* in_sizes, int n_in,
                              void* d_out, int out_size, void* d_ws, size_t ws_size,
                              hipStream_t stream)
{
    const float* xr    = (const float*)d_in[0];   // x_real  [64,16,128,128] f32
    const float* xi    = (const float*)d_in[1];   // x_imag  [64,16,128,128] f32
    const float* betas = (const float*)d_in[2];   // betas   [16384] f32
    float*       out   = (float*)d_out;           // complex64 -> interleaved f32

    const long long n    = (long long)in_sizes[0];   // 16,777,216 (divisible by EPB)
    const int       grid = (int)(n / EPB);           // 16384 blocks

    diag_exp_scale_kernel<<<grid, BLOCK, 0, stream>>>(xr, xi, betas, out);
}